// LegalGCN_4303557230748
// MI455X (gfx1250) — compile-verified
//
#include <hip/hip_runtime.h>
#include <hip/hip_bf16.h>

typedef __attribute__((ext_vector_type(2))) float v2f;
typedef __attribute__((ext_vector_type(8))) float v8f;

#define HCH 128  // hidden channels

// ---------------- degree / norm ----------------
__global__ void k_deg_init(float* deg, int N) {
    int i = blockIdx.x * blockDim.x + threadIdx.x;
    if (i < N) deg[i] = 1.0f;  // self-loop
}

__global__ void k_deg_count(const long long* __restrict__ ei, float* deg, long long E) {
    long long e = (long long)blockIdx.x * blockDim.x + threadIdx.x;
    if (e < E) atomicAdd(&deg[ei[E + e]], 1.0f);  // dst row
}

__global__ void k_deg_inv(const float* __restrict__ deg, float* dinv, int N) {
    int i = blockIdx.x * blockDim.x + threadIdx.x;
    if (i < N) dinv[i] = rsqrtf(deg[i]);  // deg >= 1 always
}

// ---------------- WMMA f32 GEMM ----------------
// h_out[N,128]   = A[N,K] @ W[128,K]^T                       (message features)
// agg_out[N,128] = h_out * dinv[i]^2 + bias[c]               (self-loop + bias, fused)
// 32 rows x 128 cols per block; 8 waves; each wave: two 16x16 tiles sharing B loads.
template <int K>
__global__ __launch_bounds__(256) void k_gemm_wmma(const float* __restrict__ A,
                                                   const float* __restrict__ W,
                                                   const float* __restrict__ dinv,
                                                   const float* __restrict__ bias,
                                                   float* __restrict__ h_out,
                                                   float* __restrict__ agg_out, int N) {
    constexpr int LDSW = K + 4;  // (K+4) % 64 == 4 for K=768,128 -> conflict-free
    __shared__ float lds[32 * LDSW];
    const int rowbase = blockIdx.x * 32;
    const int tid = threadIdx.x;

    // cooperative load of 32 x K tile of A into LDS (coalesced)
    for (int idx = tid; idx < 32 * K; idx += 256) {
        int r = idx / K, c = idx % K;
        int gr = rowbase + r;
        lds[r * LDSW + c] = (gr < N) ? A[(size_t)gr * K + c] : 0.0f;
    }
    __syncthreads();

    const int wave  = tid >> 5;          // 8 waves -> 8 column tiles of 16
    const int lane  = tid & 31;
    const int m     = lane & 15;
    const int khalf = (lane >> 4) << 1;  // 0 for lanes 0-15, 2 for lanes 16-31
    const int col   = wave * 16 + m;     // B/D column for this lane

    const float* wrow = W + (size_t)col * K;  // B[k][n] = W[col][k]
    v8f acc0 = {};  // rows rowbase+0..15
    v8f acc1 = {};  // rows rowbase+16..31

    for (int k = 0; k < K; k += 4) {
        // A 16x4 f32 layout: VGPR0 = K=k+khalf, VGPR1 = K=k+khalf+1
        v2f a0 = *(const v2f*)&lds[m * LDSW + k + khalf];
        v2f a1 = *(const v2f*)&lds[(16 + m) * LDSW + k + khalf];
        // B 4x16 f32 layout: rows k+khalf, k+khalf+1 at column `col` (shared by both tiles)
        v2f b = *(const v2f*)&wrow[k + khalf];
        acc0 = __builtin_amdgcn_wmma_f32_16x16x4_f32(false, a0, false, b,
                                                     (short)0, acc0, false, false);
        acc1 = __builtin_amdgcn_wmma_f32_16x16x4_f32(false, a1, false, b,
                                                     (short)0, acc1, false, false);
    }

    // D 16x16 f32: VGPR v holds M = v + 8*(lane>=16), N = lane&15
    const float bc = bias[col];
    const int rbase0 = rowbase + ((lane >> 4) << 3);
#pragma unroll
    for (int v = 0; v < 8; ++v) {
        int r = rbase0 + v;
        if (r < N) {
            float hv = acc0[v];
            float di = dinv[r];
            h_out[(size_t)r * HCH + col] = hv;
            agg_out[(size_t)r * HCH + col] = hv * di * di + bc;
        }
    }
    const int rbase1 = rbase0 + 16;
#pragma unroll
    for (int v = 0; v < 8; ++v) {
        int r = rbase1 + v;
        if (r < N) {
            float hv = acc1[v];
            float di = dinv[r];
            h_out[(size_t)r * HCH + col] = hv;
            agg_out[(size_t)r * HCH + col] = hv * di * di + bc;
        }
    }
}

// ---------------- aggregation: edge scatter ----------------
// one wave per edge, lane handles 4 channels: out[dst] += h[src]*dinv[src]*dinv[dst]
__global__ void k_agg_edges(const float* __restrict__ h, const long long* __restrict__ ei,
                            const float* __restrict__ dinv, float* __restrict__ out,
                            long long E) {
    long long idx = (long long)blockIdx.x * blockDim.x + threadIdx.x;
    long long e = idx >> 5;
    int lane = threadIdx.x & 31;
    if (e >= E) return;
    long long s = ei[e], d = ei[E + e];
    float nrm = dinv[s] * dinv[d];
    float4 v = *(const float4*)(h + (size_t)s * HCH + lane * 4);
    float* o = out + (size_t)d * HCH + lane * 4;
    atomicAdd(o + 0, v.x * nrm);
    atomicAdd(o + 1, v.y * nrm);
    atomicAdd(o + 2, v.z * nrm);
    atomicAdd(o + 3, v.w * nrm);
}

// ---------------- batch norm ----------------
__global__ void k_zero_stats(float* sum, float* sumsq) {
    int t = threadIdx.x;
    if (t < HCH) { sum[t] = 0.0f; sumsq[t] = 0.0f; }
}

__global__ void k_bn_stats(const float* __restrict__ x, float* sum, float* sumsq, int N) {
    __shared__ float s_sum[HCH], s_sq[HCH];
    int tid = threadIdx.x;  // 256 threads = 2 rows per step
    if (tid < HCH) { s_sum[tid] = 0.0f; s_sq[tid] = 0.0f; }
    __syncthreads();
    int c = tid & 127;
    float ls = 0.0f, lq = 0.0f;
    for (int i = blockIdx.x * 2 + (tid >> 7); i < N; i += gridDim.x * 2) {
        float v = x[(size_t)i * HCH + c];
        ls += v; lq += v * v;
    }
    atomicAdd(&s_sum[c], ls);
    atomicAdd(&s_sq[c], lq);
    __syncthreads();
    if (tid < HCH) { atomicAdd(&sum[tid], s_sum[tid]); atomicAdd(&sumsq[tid], s_sq[tid]); }
}

__global__ void k_bn_finalize(const float* sum, const float* sumsq,
                              const float* __restrict__ g, const float* __restrict__ be,
                              float* scale, float* shift, int N) {
    int c = threadIdx.x;
    if (c >= HCH) return;
    float invN = 1.0f / (float)N;
    float mean = sum[c] * invN;
    float var = sumsq[c] * invN - mean * mean;  // biased, matches jnp.var
    float sc = g[c] * rsqrtf(var + 1e-5f);
    scale[c] = sc;
    shift[c] = be[c] - mean * sc;
}

__global__ void k_bn_relu(float* __restrict__ x, const float* __restrict__ scale,
                          const float* __restrict__ shift, int N) {
    size_t t = (size_t)blockIdx.x * blockDim.x + threadIdx.x;
    if (t >= (size_t)N * HCH) return;
    int c = (int)(t & 127);
    x[t] = fmaxf(0.0f, x[t] * scale[c] + shift[c]);
}

// ---------------- head: sigmoid(a @ Wh^T + bh), one wave per node ----------------
__global__ void k_head(const float* __restrict__ a, const float* __restrict__ Wh,
                       const float* __restrict__ bh, float* __restrict__ out, int N) {
    long long idx = (long long)blockIdx.x * blockDim.x + threadIdx.x;
    int i = (int)(idx >> 5);
    int lane = threadIdx.x & 31;
    if (i >= N) return;
    float4 v = ((const float4*)(a + (size_t)i * HCH))[lane];
    float4 w = ((const float4*)Wh)[lane];
    float d = v.x * w.x + v.y * w.y + v.z * w.z + v.w * w.w;
#pragma unroll
    for (int off = 16; off > 0; off >>= 1) d += __shfl_down(d, off, 32);
    if (lane == 0) out[i] = 1.0f / (1.0f + expf(-(d + bh[0])));
}

// ---------------- driver ----------------
extern "C" void kernel_launch(void* const* d_in, const int* in_sizes, int n_in,
                              void* d_out, int out_size, void* d_ws, size_t ws_size,
                              hipStream_t stream) {
    const float*     x   = (const float*)d_in[0];
    const long long* ei  = (const long long*)d_in[1];
    const float*     W1  = (const float*)d_in[2];
    const float*     b1  = (const float*)d_in[3];
    const float*     g1  = (const float*)d_in[4];
    const float*     be1 = (const float*)d_in[5];
    const float*     W2  = (const float*)d_in[6];
    const float*     b2  = (const float*)d_in[7];
    const float*     g2  = (const float*)d_in[8];
    const float*     be2 = (const float*)d_in[9];
    const float*     Wh  = (const float*)d_in[10];
    const float*     bh  = (const float*)d_in[11];
    float* out = (float*)d_out;

    const int Din = 768;
    const int N = in_sizes[0] / Din;
    const long long E = in_sizes[1] / 2;

    char* ws = (char*)d_ws;
    float* deg   = (float*)ws; ws += (size_t)N * 4;
    float* dinv  = (float*)ws; ws += (size_t)N * 4;
    float* sum   = (float*)ws; ws += 512;
    float* sumsq = (float*)ws; ws += 512;
    float* scale = (float*)ws; ws += 512;
    float* shift = (float*)ws; ws += 512;
    float* bufA  = (float*)ws; ws += (size_t)N * HCH * 4;  // pre-aggregation h
    float* bufB  = (float*)ws; ws += (size_t)N * HCH * 4;  // aggregation acc / activations

    const int T = 256;
    int nBlkN    = (N + T - 1) / T;
    int nBlkE    = (int)((E + T - 1) / T);
    int nBlkNC   = (int)(((size_t)N * HCH + T - 1) / T);
    int nBlkE32  = (int)((E * 32 + T - 1) / T);
    int nBlkN32  = (int)(((long long)N * 32 + T - 1) / T);
    int nBlkGemm = (N + 31) / 32;

    // normalization coefficients (shared by both conv layers)
    k_deg_init<<<nBlkN, T, 0, stream>>>(deg, N);
    k_deg_count<<<nBlkE, T, 0, stream>>>(ei, deg, E);
    k_deg_inv<<<nBlkN, T, 0, stream>>>(deg, dinv, N);

    // ---- layer 1 ----  (GEMM writes h into bufA and self-loop+bias init into bufB)
    k_gemm_wmma<768><<<nBlkGemm, T, 0, stream>>>(x, W1, dinv, b1, bufA, bufB, N);
    k_agg_edges<<<nBlkE32, T, 0, stream>>>(bufA, ei, dinv, bufB, E);
    k_zero_stats<<<1, T, 0, stream>>>(sum, sumsq);
    k_bn_stats<<<1024, T, 0, stream>>>(bufB, sum, sumsq, N);
    k_bn_finalize<<<1, HCH, 0, stream>>>(sum, sumsq, g1, be1, scale, shift, N);
    k_bn_relu<<<nBlkNC, T, 0, stream>>>(bufB, scale, shift, N);

    // ---- layer 2 ----
    k_gemm_wmma<128><<<nBlkGemm, T, 0, stream>>>(bufB, W2, dinv, b2, bufA, bufB, N);
    k_agg_edges<<<nBlkE32, T, 0, stream>>>(bufA, ei, dinv, bufB, E);
    k_zero_stats<<<1, T, 0, stream>>>(sum, sumsq);
    k_bn_stats<<<1024, T, 0, stream>>>(bufB, sum, sumsq, N);
    k_bn_finalize<<<1, HCH, 0, stream>>>(sum, sumsq, g2, be2, scale, shift, N);
    k_bn_relu<<<nBlkNC, T, 0, stream>>>(bufB, scale, shift, N);

    // ---- head ----
    k_head<<<nBlkN32, T, 0, stream>>>(bufB, Wh, bh, out, N);
}